// Decoder_65498251264356
// MI455X (gfx1250) — compile-verified
//
#include <hip/hip_runtime.h>
#include <hip/hip_bf16.h>
#include <math.h>

// ---------------------------------------------------------------------------
// Decoder step (pointer-generator w/ coverage attention) for MI455X / gfx1250
// B=128, T=400, H=512, E=128, V=50000, fp32 throughout.
// GEMMs: V_WMMA_F32_16X16X4_F32, LDS-staged A/B tiles, async-to-LDS copies
// (GLOBAL_LOAD_ASYNC_TO_LDS_B128 + s_wait_asynccnt) for the staging.
// ---------------------------------------------------------------------------

#define B_  128
#define T_  400
#define H_  512
#define E_  128
#define V_  50000
#define EPS_ 1e-12f
#define KC_ 64          // K-chunk staged in LDS (all K here are multiples of 64)

typedef __attribute__((ext_vector_type(2))) float v2f;
typedef __attribute__((ext_vector_type(8))) float v8f;
typedef __attribute__((ext_vector_type(4))) int   v4i;

#if defined(__has_builtin)
#  if __has_builtin(__builtin_amdgcn_global_load_async_to_lds_b128)
#    define HAVE_ASYNC_LDS 1
#  endif
#  if __has_builtin(__builtin_amdgcn_s_wait_asynccnt)
#    define HAVE_WAIT_ASYNC_BUILTIN 1
#  endif
#endif

typedef v4i __attribute__((address_space(1)))* gv4i_p;   // global (device) ptr
typedef v4i __attribute__((address_space(3)))* lv4i_p;   // LDS (shared) ptr

__device__ __forceinline__ void copy16_to_lds(const float* __restrict__ g,
                                              float* __restrict__ l) {
#ifdef HAVE_ASYNC_LDS
  __builtin_amdgcn_global_load_async_to_lds_b128((gv4i_p)(void*)g,
                                                 (lv4i_p)(void*)l, 0, 0);
#else
  *(float4*)l = *(const float4*)g;
#endif
}

__device__ __forceinline__ void staging_wait() {
#ifdef HAVE_ASYNC_LDS
#  ifdef HAVE_WAIT_ASYNC_BUILTIN
  __builtin_amdgcn_s_wait_asynccnt(0);
#  else
  asm volatile("s_wait_asynccnt 0x0" ::: "memory");
#  endif
#endif
}

__device__ __forceinline__ float sigmoidf_(float x) { return 1.0f / (1.0f + __expf(-x)); }

// ---------------------------------------------------------------------------
// GEMM: C[128, N] = A[128, K] @ W[N, K]^T + bias   (+= C if acc)
// Block = 8 waves = 128 output columns. Per K-chunk (64):
//   - cooperatively stage A[128,64] and W[128 rows, 64] into LDS (coalesced,
//     async-to-LDS),
//   - each wave runs 16 k-steps of 8 WMMA tiles (full M=128) fed by ds_load.
// A-matrix lane layout (16x4 f32): v0 = A[m, k+2*half], v1 = A[m, k+2*half+1]
// B-matrix lane layout (4x16 f32): v0 = W[n, k+2*half], v1 = W[n, k+2*half+1]
// C/D layout (16x16 f32): vgpr v holds row (v + 8*half), col = lane&15.
// ---------------------------------------------------------------------------
__global__ void __launch_bounds__(256)
gemm128_wmma(const float* __restrict__ A, const float* __restrict__ W,
             const float* __restrict__ bias, float* __restrict__ C,
             int N, int K, int acc) {
  __shared__ float sA[128 * KC_];   // 32 KB
  __shared__ float sB[128 * KC_];   // 32 KB

  const int tid  = threadIdx.x;
  const int lane = tid & 31;
  const int wave = tid >> 5;
  const int nbase = blockIdx.x * 128;          // first column of this block
  const int col0  = nbase + wave * 16;         // this wave's 16-col tile

  const int half = lane >> 4;                  // 0 or 1
  const int l16  = lane & 15;
  const int kb   = 2 * half;

  v8f accv[8];
#pragma unroll
  for (int mt = 0; mt < 8; ++mt) accv[mt] = (v8f)0.0f;

  for (int k0 = 0; k0 < K; k0 += KC_) {
    // ---- stage A[0:128, k0:k0+64] and W[nbase:nbase+128, k0:k0+64] ----
#pragma unroll
    for (int it = 0; it < 8; ++it) {           // 2048 float4 / 256 threads
      const int i   = it * 256 + tid;
      const int row = i >> 4;                  // 16 float4 per row
      const int c4  = (i & 15) * 4;
      copy16_to_lds(A + (size_t)row * K + k0 + c4, &sA[row * KC_ + c4]);
      const int wrow = min(nbase + row, N - 1);
      copy16_to_lds(W + (size_t)wrow * K + k0 + c4, &sB[row * KC_ + c4]);
    }
    staging_wait();
    __syncthreads();

    // ---- compute: 16 k-steps x 8 M-tiles ----
#pragma unroll 4
    for (int kk = 0; kk < KC_; kk += 4) {
      const v2f b = *(const v2f*)&sB[(wave * 16 + l16) * KC_ + kk + kb];
      v2f a[8];
#pragma unroll
      for (int mt = 0; mt < 8; ++mt)
        a[mt] = *(const v2f*)&sA[(mt * 16 + l16) * KC_ + kk + kb];
#pragma unroll
      for (int mt = 0; mt < 8; ++mt)
        accv[mt] = __builtin_amdgcn_wmma_f32_16x16x4_f32(
            false, a[mt], false, b, (short)0, accv[mt], false, false);
    }
    __syncthreads();
  }

  // ---- epilogue ----
  const int n = col0 + l16;
  if (n < N) {
    const float bv = bias ? bias[n] : 0.0f;
#pragma unroll
    for (int mt = 0; mt < 8; ++mt) {
#pragma unroll
      for (int v = 0; v < 8; ++v) {
        const int m = mt * 16 + v + 8 * half;
        float val = accv[mt][v] + bv;
        size_t idx = (size_t)m * (size_t)N + (size_t)n;
        if (acc) val += C[idx];
        C[idx] = val;
      }
    }
  }
}

// ---------------------------------------------------------------------------
// xcat[b, 0:2H] = c_t_1[b], xcat[b, 2H:2H+E] = emb_table[y_t_1[b]]
// ---------------------------------------------------------------------------
__global__ void build_xcat(const float* __restrict__ c_t_1,
                           const int* __restrict__ y,
                           const float* __restrict__ emb,
                           float* __restrict__ xcat) {
  int idx = blockIdx.x * blockDim.x + threadIdx.x;
  const int KCc = 2 * H_ + E_;   // 1152
  if (idx >= B_ * KCc) return;
  int b = idx / KCc, k = idx % KCc;
  xcat[idx] = (k < 2 * H_) ? c_t_1[b * (2 * H_) + k]
                           : emb[(size_t)y[b] * E_ + (k - 2 * H_)];
}

// ---------------------------------------------------------------------------
// LSTM elementwise: gates [B,4H] (i,f,g,o) -> h, c, with fan-out.
// ---------------------------------------------------------------------------
__global__ void lstm_cell(const float* __restrict__ gates,
                          const float* __restrict__ c0,
                          float* __restrict__ h_out, float* __restrict__ c_out,
                          float* __restrict__ s_t_hat, float* __restrict__ hc_ct) {
  int idx = blockIdx.x * blockDim.x + threadIdx.x;
  if (idx >= B_ * H_) return;
  int b = idx / H_, j = idx % H_;
  const float* g = gates + (size_t)b * 4 * H_;
  float ig = sigmoidf_(g[j]);
  float fg = sigmoidf_(g[H_ + j]);
  float gg = tanhf(g[2 * H_ + j]);
  float og = sigmoidf_(g[3 * H_ + j]);
  float c = fg * c0[idx] + ig * gg;
  float h = og * tanhf(c);
  h_out[idx] = h;
  c_out[idx] = c;
  s_t_hat[(size_t)b * 2 * H_ + j] = h;
  s_t_hat[(size_t)b * 2 * H_ + H_ + j] = c;
  hc_ct[(size_t)b * 3 * H_ + j] = h;
}

// ---------------------------------------------------------------------------
// Coverage attention. One block per batch row. 8 waves x 32 lanes.
// ---------------------------------------------------------------------------
__global__ void attention_kernel(const float* __restrict__ enc_feat,
                                 const float* __restrict__ enc_out,
                                 const float* __restrict__ dec_fea,
                                 const float* __restrict__ W_c,
                                 const float* __restrict__ v_w,
                                 const float* __restrict__ mask,
                                 const float* __restrict__ coverage,
                                 float* __restrict__ attn_out,
                                 float* __restrict__ cov_next,
                                 float* __restrict__ c_t_out,
                                 float* __restrict__ hc_ct) {
  __shared__ float s_dec[2 * H_];
  __shared__ float s_wc[2 * H_];
  __shared__ float s_v[2 * H_];
  __shared__ float s_sc[T_];
  __shared__ float s_red[256];

  const int b = blockIdx.x;
  const int tid = threadIdx.x;
  const int lane = tid & 31;
  const int wave = tid >> 5;

  for (int i = tid; i < 2 * H_; i += 256) {
    s_dec[i] = dec_fea[(size_t)b * 2 * H_ + i];
    s_wc[i]  = W_c[i];
    s_v[i]   = v_w[i];
  }
  __syncthreads();

  for (int t = wave; t < T_; t += 8) {
    const float cov = coverage[b * T_ + t];
    const float* __restrict__ ef = enc_feat + ((size_t)b * T_ + t) * (2 * H_);
    float p = 0.0f;
    for (int n = lane; n < 2 * H_; n += 32)
      p += s_v[n] * tanhf(ef[n] + s_dec[n] + cov * s_wc[n]);
#pragma unroll
    for (int off = 16; off > 0; off >>= 1) p += __shfl_xor(p, off, 32);
    if (lane == 0) s_sc[t] = p;
  }
  __syncthreads();

  float m = -INFINITY;
  for (int t = tid; t < T_; t += 256) m = fmaxf(m, s_sc[t]);
  s_red[tid] = m; __syncthreads();
  for (int s = 128; s > 0; s >>= 1) {
    if (tid < s) s_red[tid] = fmaxf(s_red[tid], s_red[tid + s]);
    __syncthreads();
  }
  const float rowmax = s_red[0];
  __syncthreads();

  float lsum = 0.0f;
  for (int t = tid; t < T_; t += 256) {
    float e = __expf(s_sc[t] - rowmax);
    s_sc[t] = e;
    lsum += e;
  }
  s_red[tid] = lsum; __syncthreads();
  for (int s = 128; s > 0; s >>= 1) {
    if (tid < s) s_red[tid] += s_red[tid + s];
    __syncthreads();
  }
  const float inv1 = 1.0f / s_red[0];
  __syncthreads();

  float lsum2 = 0.0f;
  for (int t = tid; t < T_; t += 256) {
    float a = s_sc[t] * inv1 * mask[b * T_ + t];
    s_sc[t] = a;
    lsum2 += a;
  }
  s_red[tid] = lsum2; __syncthreads();
  for (int s = 128; s > 0; s >>= 1) {
    if (tid < s) s_red[tid] += s_red[tid + s];
    __syncthreads();
  }
  const float inv2 = 1.0f / (s_red[0] + EPS_);
  __syncthreads();

  for (int t = tid; t < T_; t += 256) {
    float a = s_sc[t] * inv2;
    s_sc[t] = a;
    attn_out[b * T_ + t] = a;
    cov_next[b * T_ + t] = coverage[b * T_ + t] + a;
  }
  __syncthreads();

  for (int n = tid; n < 2 * H_; n += 256) {
    const float* __restrict__ eo = enc_out + (size_t)b * T_ * (2 * H_) + n;
    float acc = 0.0f;
    for (int t = 0; t < T_; ++t) acc += s_sc[t] * eo[(size_t)t * (2 * H_)];
    c_t_out[(size_t)b * 2 * H_ + n] = acc;
    hc_ct[(size_t)b * 3 * H_ + H_ + n] = acc;
  }
}

// ---------------------------------------------------------------------------
// p_gen[b] = sigmoid(W_pg . [c_t, s_t_hat, x] + b_pg)
// ---------------------------------------------------------------------------
__global__ void pgen_kernel(const float* __restrict__ c_t,
                            const float* __restrict__ s_t_hat,
                            const float* __restrict__ x,
                            const float* __restrict__ W_pg,
                            const float* __restrict__ b_pg,
                            float* __restrict__ pgen) {
  __shared__ float s_red[256];
  const int b = blockIdx.x, tid = threadIdx.x;
  const int KCc = 4 * H_ + E_;   // 2176
  float acc = 0.0f;
  for (int k = tid; k < KCc; k += 256) {
    float v;
    if (k < 2 * H_)            v = c_t[(size_t)b * 2 * H_ + k];
    else if (k < 4 * H_)       v = s_t_hat[(size_t)b * 2 * H_ + (k - 2 * H_)];
    else                       v = x[(size_t)b * E_ + (k - 4 * H_)];
    acc += v * W_pg[k];
  }
  s_red[tid] = acc; __syncthreads();
  for (int s = 128; s > 0; s >>= 1) {
    if (tid < s) s_red[tid] += s_red[tid + s];
    __syncthreads();
  }
  if (tid == 0) pgen[b] = sigmoidf_(s_red[0] + b_pg[0]);
}

// ---------------------------------------------------------------------------
// vocab softmax pass 1: per-row max and sum(exp)
// ---------------------------------------------------------------------------
__global__ void softmax_stats(const float* __restrict__ logits,
                              float* __restrict__ rowmax,
                              float* __restrict__ rowsum) {
  __shared__ float s_red[512];
  const int b = blockIdx.x, tid = threadIdx.x;
  const float* __restrict__ row = logits + (size_t)b * V_;
  float m = -INFINITY;
  for (int v = tid; v < V_; v += 512) m = fmaxf(m, row[v]);
  s_red[tid] = m; __syncthreads();
  for (int s = 256; s > 0; s >>= 1) {
    if (tid < s) s_red[tid] = fmaxf(s_red[tid], s_red[tid + s]);
    __syncthreads();
  }
  const float rm = s_red[0];
  __syncthreads();
  float acc = 0.0f;
  for (int v = tid; v < V_; v += 512) acc += __expf(row[v] - rm);
  s_red[tid] = acc; __syncthreads();
  for (int s = 256; s > 0; s >>= 1) {
    if (tid < s) s_red[tid] += s_red[tid + s];
    __syncthreads();
  }
  if (tid == 0) { rowmax[b] = rm; rowsum[b] = s_red[0]; }
}

__global__ void vocab_dist(const float* __restrict__ logits,
                           const float* __restrict__ rowmax,
                           const float* __restrict__ rowsum,
                           const float* __restrict__ pgen,
                           float* __restrict__ final) {
  size_t idx = (size_t)blockIdx.x * blockDim.x + threadIdx.x;
  if (idx >= (size_t)B_ * V_) return;
  int b = (int)(idx / V_);
  final[idx] = pgen[b] * __expf(logits[idx] - rowmax[b]) / rowsum[b];
}

__global__ void scatter_add(float* __restrict__ final,
                            const int* __restrict__ enc_vocab,
                            const float* __restrict__ attn,
                            const float* __restrict__ pgen) {
  int idx = blockIdx.x * blockDim.x + threadIdx.x;
  if (idx >= B_ * T_) return;
  int b = idx / T_;
  atomicAdd(final + (size_t)b * V_ + enc_vocab[idx], (1.0f - pgen[b]) * attn[idx]);
}

// ---------------------------------------------------------------------------
extern "C" void kernel_launch(void* const* d_in, const int* in_sizes, int n_in,
                              void* d_out, int out_size, void* d_ws, size_t ws_size,
                              hipStream_t stream) {
  const int*   y_t_1    = (const int*)  d_in[0];
  const float* h0       = (const float*)d_in[1];
  const float* c0       = (const float*)d_in[2];
  const float* c_t_1    = (const float*)d_in[3];
  const float* enc_out  = (const float*)d_in[4];
  const float* enc_feat = (const float*)d_in[5];
  const float* mask     = (const float*)d_in[6];
  const int*   enc_voc  = (const int*)  d_in[7];
  const float* coverage = (const float*)d_in[8];
  const float* emb      = (const float*)d_in[9];
  const float* W_c      = (const float*)d_in[10];
  const float* W_dp     = (const float*)d_in[11];
  const float* b_dp     = (const float*)d_in[12];
  const float* v_w      = (const float*)d_in[13];
  const float* W_xc     = (const float*)d_in[14];
  const float* b_xc     = (const float*)d_in[15];
  const float* W_ih     = (const float*)d_in[16];
  const float* W_hh     = (const float*)d_in[17];
  const float* b_ih     = (const float*)d_in[18];
  const float* b_hh     = (const float*)d_in[19];
  const float* W_pg     = (const float*)d_in[20];
  const float* b_pg     = (const float*)d_in[21];
  const float* W_o1     = (const float*)d_in[22];
  const float* b_o1     = (const float*)d_in[23];
  const float* W_o2     = (const float*)d_in[24];
  const float* b_o2     = (const float*)d_in[25];

  float* out       = (float*)d_out;
  float* o_final   = out;                                   // [B,V]
  float* o_h       = out + (size_t)B_ * V_;                 // [B,H]
  float* o_c       = o_h + (size_t)B_ * H_;                 // [B,H]
  float* o_ct      = o_c + (size_t)B_ * H_;                 // [B,2H]
  float* o_attn    = o_ct + (size_t)B_ * 2 * H_;            // [B,T]
  float* o_pgen    = o_attn + (size_t)B_ * T_;              // [B,1]
  float* o_cov     = o_pgen + (size_t)B_;                   // [B,T]

  float* ws       = (float*)d_ws;
  float* w_xcat   = ws;                                     // [B, 2H+E]
  float* w_x      = w_xcat + (size_t)B_ * (2 * H_ + E_);    // [B, E]
  float* w_gates  = w_x + (size_t)B_ * E_;                  // [B, 4H]
  float* w_shat   = w_gates + (size_t)B_ * 4 * H_;          // [B, 2H]
  float* w_decf   = w_shat + (size_t)B_ * 2 * H_;           // [B, 2H]
  float* w_hcct   = w_decf + (size_t)B_ * 2 * H_;           // [B, 3H]
  float* w_out1   = w_hcct + (size_t)B_ * 3 * H_;           // [B, H]
  float* w_logit  = w_out1 + (size_t)B_ * H_;               // [B, V]
  float* w_rmax   = w_logit + (size_t)B_ * V_;              // [B]
  float* w_rsum   = w_rmax + B_;                            // [B]

  #define GEMM_GRID(N) (((N) + 127) / 128)

  build_xcat<<<(B_ * (2 * H_ + E_) + 255) / 256, 256, 0, stream>>>(c_t_1, y_t_1, emb, w_xcat);

  // x = xcat @ W_xc.T + b_xc              (N=128,  K=1152)
  gemm128_wmma<<<GEMM_GRID(E_), 256, 0, stream>>>(w_xcat, W_xc, b_xc, w_x, E_, 2 * H_ + E_, 0);

  // gates = x @ W_ih.T + b_ih             (N=2048, K=128)
  gemm128_wmma<<<GEMM_GRID(4 * H_), 256, 0, stream>>>(w_x, W_ih, b_ih, w_gates, 4 * H_, E_, 0);
  // gates += h0 @ W_hh.T + b_hh           (N=2048, K=512)
  gemm128_wmma<<<GEMM_GRID(4 * H_), 256, 0, stream>>>(h0, W_hh, b_hh, w_gates, 4 * H_, H_, 1);

  lstm_cell<<<(B_ * H_ + 255) / 256, 256, 0, stream>>>(w_gates, c0, o_h, o_c, w_shat, w_hcct);

  // dec_fea = s_t_hat @ W_dp.T + b_dp     (N=1024, K=1024)
  gemm128_wmma<<<GEMM_GRID(2 * H_), 256, 0, stream>>>(w_shat, W_dp, b_dp, w_decf, 2 * H_, 2 * H_, 0);

  attention_kernel<<<B_, 256, 0, stream>>>(enc_feat, enc_out, w_decf, W_c, v_w, mask,
                                           coverage, o_attn, o_cov, o_ct, w_hcct);

  pgen_kernel<<<B_, 256, 0, stream>>>(o_ct, w_shat, w_x, W_pg, b_pg, o_pgen);

  // out1 = [h, c_t] @ W_o1.T + b_o1       (N=512,  K=1536)
  gemm128_wmma<<<GEMM_GRID(H_), 256, 0, stream>>>(w_hcct, W_o1, b_o1, w_out1, H_, 3 * H_, 0);

  // logits = out1 @ W_o2.T + b_o2         (N=50000, K=512) -- dominant GEMM
  gemm128_wmma<<<GEMM_GRID(V_), 256, 0, stream>>>(w_out1, W_o2, b_o2, w_logit, V_, H_, 0);

  softmax_stats<<<B_, 512, 0, stream>>>(w_logit, w_rmax, w_rsum);
  vocab_dist<<<(int)(((size_t)B_ * V_ + 255) / 256), 256, 0, stream>>>(w_logit, w_rmax, w_rsum, o_pgen, o_final);
  scatter_add<<<(B_ * T_ + 255) / 256, 256, 0, stream>>>(o_final, enc_voc, o_attn, o_pgen);
}